// FeatureSim_1786706395424
// MI455X (gfx1250) — compile-verified
//
#include <hip/hip_runtime.h>
#include <hip/hip_bf16.h>
#include <math.h>

typedef __attribute__((ext_vector_type(2))) float v2f;
typedef __attribute__((ext_vector_type(4))) float v4f;
typedef __attribute__((ext_vector_type(8))) float v8f;
typedef __attribute__((ext_vector_type(4))) int   i4;

#define BQ    8
#define LQ    1024
#define NF    11
#define FQ    16
#define ITILE 16
#define SROW  1026   // padded LDS row stride (floats): stride%64==2 -> WMMA A-loads conflict-free

#ifndef TRY_ASYNC_LDS
#define TRY_ASYNC_LDS 1
#endif
#if TRY_ASYNC_LDS && __has_builtin(__builtin_amdgcn_global_load_async_to_lds_b128)
#define USE_ASYNC_LDS 1
typedef __attribute__((address_space(1))) i4* as1_i4p;
typedef __attribute__((address_space(3))) i4* as3_i4p;
#endif

__global__ __launch_bounds__(512) void featsim_kernel(
    const float* __restrict__ x, const int* __restrict__ xlen,
    const float* __restrict__ fi, float* __restrict__ out)
{
    __shared__ float xi[ITILE * FQ];        // i-tile rows (stride 16)
    __shared__ float S[ITILE * SROW];       // logits -> exp tile, 16 x 1024 (padded)
    __shared__ float part[16 * ITILE];      // per-wave WMMA partial row sums

    const int t    = threadIdx.x;
    const int b    = blockIdx.x >> 6;             // 64 i-tiles per batch
    const int i0   = (blockIdx.x & 63) * ITILE;
    const float* xb = x + ((size_t)b * LQ * FQ);

    // ---- stage i-tile (16 rows x 64B) into LDS via async copy (ASYNCcnt path) ----
#ifdef USE_ASYNC_LDS
    if (t < (ITILE * FQ) / 4) {                   // 64 x b128 transfers
        const float* gp = xb + (size_t)i0 * FQ + t * 4;
        float*       lp = &xi[t * 4];
        __builtin_amdgcn_global_load_async_to_lds_b128(
            (as1_i4p)((void*)gp), (as3_i4p)((void*)lp), 0, 0);
    }
#if __has_builtin(__builtin_amdgcn_s_wait_asynccnt)
    __builtin_amdgcn_s_wait_asynccnt(0);
#else
    asm volatile("s_wait_asynccnt 0" ::: "memory");
#endif
#else
    if (t < ITILE * FQ) xi[t] = xb[(size_t)i0 * FQ + t];
#endif

    float w[NF];
#pragma unroll
    for (int k = 0; k < NF; ++k) w[k] = fi[k];
    const int len = xlen[b];

    __syncthreads();

    // ---- pass 1: masked logits for columns j = t and t+512 against all 16 rows ----
    {
        const int j0 = t, j1 = t + 512;
        const float* xr0 = xb + (size_t)j0 * FQ;
        const float* xr1 = xb + (size_t)j1 * FQ;
        v4f a0 = *(const v4f*)(xr0), a1 = *(const v4f*)(xr0 + 4), a2 = *(const v4f*)(xr0 + 8);
        v4f b0 = *(const v4f*)(xr1), b1 = *(const v4f*)(xr1 + 4), b2 = *(const v4f*)(xr1 + 8);
        float xa[12], xc[12];
#pragma unroll
        for (int k = 0; k < 4; ++k) {
            xa[k] = a0[k]; xa[k + 4] = a1[k]; xa[k + 8] = a2[k];
            xc[k] = b0[k]; xc[k + 4] = b1[k]; xc[k + 8] = b2[k];
        }
        const bool v0 = j0 < len, v1 = j1 < len;
        const float ninf = -__builtin_inff();
#pragma unroll
        for (int ii = 0; ii < ITILE; ++ii) {
            v4f c0 = *(const v4f*)&xi[ii * FQ];
            v4f c1 = *(const v4f*)&xi[ii * FQ + 4];
            v4f c2 = *(const v4f*)&xi[ii * FQ + 8];
            float ci[12];
#pragma unroll
            for (int k = 0; k < 4; ++k) { ci[k] = c0[k]; ci[k + 4] = c1[k]; ci[k + 8] = c2[k]; }
            float s0 = 0.f, s1 = 0.f;
#pragma unroll
            for (int k = 0; k < NF; ++k) {
                s0 += w[k] * __builtin_fabsf(xa[k] - ci[k]);
                s1 += w[k] * __builtin_fabsf(xc[k] - ci[k]);
            }
            S[ii * SROW + j0] = v0 ? s0 : ninf;
            S[ii * SROW + j1] = v1 ? s1 : ninf;
        }
    }
    __syncthreads();

    const int wv = t >> 5, lane = t & 31;   // 16 waves, wave wv owns row wv

    // ---- pass 2a: row max (shuffle butterfly), exp in place ----
    {
        float* row = &S[wv * SROW];
        float m = -__builtin_inff();
#pragma unroll
        for (int q = 0; q < 32; ++q) m = fmaxf(m, row[lane + q * 32]);
#pragma unroll
        for (int off = 16; off > 0; off >>= 1) m = fmaxf(m, __shfl_xor(m, off, 32));
        const bool live = m > -__builtin_inff();   // len>=1 -> always true; NaN guard
#pragma unroll
        for (int q = 0; q < 32; ++q) {
            float e = live ? __expf(row[lane + q * 32] - m) : 0.f;
            row[lane + q * 32] = e;
        }
    }
    __syncthreads();

    // ---- pass 2b: WMMA row-sums. Each wave reduces its 64-col slab of the exp tile:
    //      D = sum_c A_c(16x4) * Ones(4x16)  => D[m,n] = rowsum_m (replicated over n)
    {
        const v2f bones = {1.f, 1.f};
        v8f acc = {};
        const int row16 = lane & 15;
        const int chi   = (lane >> 4) << 1;                 // A layout: lanes>=16 hold K=2,3
        const float* base = &S[row16 * SROW + wv * 64 + chi];
#pragma unroll
        for (int c = 0; c < 16; ++c) {
            v2f a = *(const v2f*)(base + c * 4);
            acc = __builtin_amdgcn_wmma_f32_16x16x4_f32(
                false, a, false, bones, (short)0, acc, false, false);
        }
        // lane 0 holds D rows 0..7 (col 0), lane 16 holds rows 8..15
        if ((lane & 15) == 0) {
            const int mbase = (lane >> 4) * 8;
#pragma unroll
            for (int r = 0; r < 8; ++r) part[wv * 16 + mbase + r] = acc[r];
        }
    }
    __syncthreads();

    // ---- pass 2c: combine 16 partials per row, scale, coalesced store ----
    {
        float p = (lane < 16) ? part[lane * 16 + wv] : 0.f;
#pragma unroll
        for (int off = 16; off > 0; off >>= 1) p += __shfl_xor(p, off, 32);
        const float inv = (p > 0.f) ? (1.0f / p) : 0.f;     // NaN->0 semantics
        const float* row = &S[wv * SROW];
        float* op = out + ((size_t)(b * LQ + i0 + wv)) * LQ;
#pragma unroll
        for (int q = 0; q < 32; ++q) op[lane + q * 32] = row[lane + q * 32] * inv;
    }
}

extern "C" void kernel_launch(void* const* d_in, const int* in_sizes, int n_in,
                              void* d_out, int out_size, void* d_ws, size_t ws_size,
                              hipStream_t stream) {
    const float* x   = (const float*)d_in[0];
    const int*   xl  = (const int*)d_in[1];
    const float* fi  = (const float*)d_in[2];
    float*       out = (float*)d_out;
    (void)in_sizes; (void)n_in; (void)out_size; (void)d_ws; (void)ws_size;
    dim3 grid(BQ * (LQ / ITILE));   // 512 blocks: one per (batch, 16-row i-tile)
    dim3 block(512);                // 16 wave32
    featsim_kernel<<<grid, block, 0, stream>>>(x, xl, fi, out);
}